// eGAUSSpPyTorchV3_fixed_17489106829504
// MI455X (gfx1250) — compile-verified
//
#include <hip/hip_runtime.h>
#include <math.h>

#define D 64
#define C 128
#define NMAXF 100.0f
#define GMAX 0.9f
#define KJOIN 0.9f

typedef __attribute__((ext_vector_type(2))) float v2f;
typedef __attribute__((ext_vector_type(8))) float v8f;

// ---------------------------------------------------------------------------
// Kernel 1: per-cluster squared distance + Gamma.
// One 64-thread block per cluster. Mahalanobis path does a cooperative 64x64
// Cholesky in LDS and two triangular solves (pinv of SPD == inv).
// ---------------------------------------------------------------------------
__global__ __launch_bounds__(64) void k_gamma(
    const float* __restrict__ z, const float* __restrict__ mu,
    const float* __restrict__ S, const float* __restrict__ n,
    const int* __restrict__ c_active, float* __restrict__ Gamma)
{
  __shared__ float A[D][D + 1];
  __shared__ float ev[D];
  __shared__ float red[D];
  __shared__ float yv[D];
  const int c = blockIdx.x;
  const int tid = threadIdx.x;
  const float nc = n[c];

  ev[tid] = z[tid] - mu[c * D + tid];
  __syncthreads();
  red[tid] = ev[tid] * ev[tid];
  __syncthreads();
  for (int s = 32; s > 0; s >>= 1) { if (tid < s) red[tid] += red[tid + s]; __syncthreads(); }
  float d2 = red[0];
  __syncthreads();

  if (nc >= NMAXF) {
    for (int idx = tid; idx < D * D; idx += 64)
      A[idx >> 6][idx & 63] = S[(size_t)c * D * D + idx] / nc;
    __syncthreads();
    // right-looking Cholesky, thread = row
    for (int j = 0; j < D; ++j) {
      if (tid == 0) { float dv = A[j][j]; A[j][j] = sqrtf(fmaxf(dv, 1e-30f)); }
      __syncthreads();
      if (tid > j) A[tid][j] /= A[j][j];
      __syncthreads();
      if (tid > j) {
        float l = A[tid][j];
        for (int kk = j + 1; kk <= tid; ++kk) A[tid][kk] -= l * A[kk][j];
      }
      __syncthreads();
    }
    // forward solve L y = e
    yv[tid] = ev[tid];
    __syncthreads();
    for (int j = 0; j < D; ++j) {
      if (tid == j) yv[j] /= A[j][j];
      __syncthreads();
      if (tid > j) yv[tid] -= A[tid][j] * yv[j];
      __syncthreads();
    }
    // back solve L^T x = y
    for (int j = D - 1; j >= 0; --j) {
      if (tid == j) yv[j] /= A[j][j];
      __syncthreads();
      if (tid < j) yv[tid] -= A[j][tid] * yv[j];
      __syncthreads();
    }
    red[tid] = ev[tid] * yv[tid];
    __syncthreads();
    for (int s = 32; s > 0; s >>= 1) { if (tid < s) red[tid] += red[tid + s]; __syncthreads(); }
    d2 = red[0];
  }
  if (tid == 0) Gamma[c] = (c < c_active[0]) ? expf(-d2) : 0.0f;
}

// ---------------------------------------------------------------------------
// Kernel 2: argmax(Gamma) (first-index tie-break), copy mu/n into workspace,
// apply the incr/add branch, compute sel mask. Single 128-thread block.
// S is updated *virtually*: upd[] + ej[] describe the delta for one cluster.
// ---------------------------------------------------------------------------
__global__ __launch_bounds__(128) void k_select(
    const float* __restrict__ z, const float* __restrict__ mu,
    const float* __restrict__ n, const int* __restrict__ c_active,
    float* __restrict__ Gamma, float* __restrict__ mu2, float* __restrict__ n2,
    int* __restrict__ active2, int* __restrict__ sel,
    float* __restrict__ ej, int* __restrict__ upd)
{
  __shared__ float gv[C];
  __shared__ int gi[C];
  __shared__ int jbest_s;
  __shared__ float gmax_s;
  const int tid = threadIdx.x;
  const int ca = c_active[0];

  gv[tid] = Gamma[tid]; gi[tid] = tid;
  __syncthreads();
  for (int s = 64; s > 0; s >>= 1) {
    if (tid < s) {
      float a = gv[tid], b = gv[tid + s];
      if (b > a || (b == a && gi[tid + s] < gi[tid])) { gv[tid] = b; gi[tid] = gi[tid + s]; }
    }
    __syncthreads();
  }
  if (tid == 0) { jbest_s = gi[0]; gmax_s = gv[0]; }

  n2[tid] = n[tid];
  active2[tid] = (tid < ca) ? 1 : 0;
  for (int idx = tid; idx < C * D; idx += 128) mu2[idx] = mu[idx];
  __syncthreads();

  const int jb = jbest_s;
  const float gmax = gmax_s;
  if (gmax > GMAX) {            // incr branch
    if (tid < D) {
      float e = z[tid] - mu[jb * D + tid];
      ej[tid] = e;
      mu2[jb * D + tid] = mu[jb * D + tid] + e / (1.0f + n[jb]);
    }
    if (tid == 0) { n2[jb] = n[jb] + 1.0f; upd[0] = 0; upd[1] = jb; }
  } else {                      // add branch
    if (tid < D) { mu2[ca * D + tid] = z[tid]; ej[tid] = 0.0f; }
    if (tid == 0) { n2[ca] = 1.0f; Gamma[ca] = 1.0f; active2[ca] = 1; upd[0] = 1; upd[1] = ca; }
  }
  __syncthreads();
  sel[tid] = (Gamma[tid] > GMAX * 0.25f && active2[tid]) ? 1 : 0;
}

// ---------------------------------------------------------------------------
// Kernel 3: det(S'_c / n'_c) via Cholesky (log-domain). V_FACTOR cancels in
// kappa so raw determinants are stored. One 64-thread block per cluster.
// ---------------------------------------------------------------------------
__global__ __launch_bounds__(64) void k_detii(
    const float* __restrict__ S, const float* __restrict__ n2,
    const float* __restrict__ ej, const int* __restrict__ upd,
    float* __restrict__ det_ii)
{
  __shared__ float A[D][D + 1];
  __shared__ float red[D];
  const int c = blockIdx.x;
  const int tid = threadIdx.x;
  const int ut = upd[0], ui = upd[1];
  const float nc = n2[c];

  for (int idx = tid; idx < D * D; idx += 64) {
    int r = idx >> 6, cc = idx & 63;
    float v = S[(size_t)c * D * D + idx];
    if (c == ui) v = (ut == 0) ? (v + ej[r] * ej[cc]) : ((r == cc) ? 1.0f : 0.0f);
    A[r][cc] = v / nc;
  }
  __syncthreads();
  for (int j = 0; j < D; ++j) {
    if (tid == 0) { float dv = A[j][j]; A[j][j] = sqrtf(fmaxf(dv, 1e-30f)); }
    __syncthreads();
    if (tid > j) A[tid][j] /= A[j][j];
    __syncthreads();
    if (tid > j) {
      float l = A[tid][j];
      for (int kk = j + 1; kk <= tid; ++kk) A[tid][kk] -= l * A[kk][j];
    }
    __syncthreads();
  }
  red[tid] = 2.0f * logf(A[tid][tid]);
  __syncthreads();
  for (int s = 32; s > 0; s >>= 1) { if (tid < s) red[tid] += red[tid + s]; __syncthreads(); }
  if (tid == 0) det_ii[c] = expf(red[0]);
}

// ---------------------------------------------------------------------------
// Kernel 4: per-pair Sigma build + blocked Cholesky (panel=16) with
// V_WMMA_F32_16X16X4_F32 trailing updates (full-fp32 WMMA path). One wave32
// workgroup per flat pair index; non-valid pairs write +inf. EXEC is all-ones
// at every wmma (no divergence after the early return).
//
// Fragment layouts per CDNA5 ISA 7.12.2:
//   A 16x4 f32 : lane l -> row l%16, holds K = {0,1} (lanes 0-15) / {2,3}
//                (lanes 16-31) in VGPR0/1  -> v2f {a.x, a.y}
//   B 4x16 f32 : mirrored (lane l -> col l%16, same K split), so loading
//                P_bj with the A-gather yields exactly P_bj^T as the B operand.
//   C/D 16x16  : VGPR v, lane l -> row 16*bi + v + (l<16?0:8), col l%16
// ---------------------------------------------------------------------------
__global__ __launch_bounds__(32) void k_pair(
    const float* __restrict__ S, const float* __restrict__ mu2,
    const float* __restrict__ n2, const int* __restrict__ sel,
    const float* __restrict__ ej, const int* __restrict__ upd,
    const float* __restrict__ det_ii, float* __restrict__ kappa)
{
  __shared__ float Sig[D][D + 1];
  __shared__ float mui[D], muj[D], muij[D], ejs[D];
  __shared__ float red[32];
  __shared__ float sdiag_s;
  __shared__ int fail_s;

  const int f = blockIdx.x;
  const int i = f >> 7, j = f & 127;
  const int lane = threadIdx.x;
  if (!(i < j) || !sel[i] || !sel[j]) {
    if (lane == 0) kappa[f] = __builtin_inff();
    return;
  }
  __builtin_prefetch(&S[(size_t)i * D * D], 0, 0);
  __builtin_prefetch(&S[(size_t)j * D * D], 0, 0);

  const int ut = upd[0], ui = upd[1];
  const float ni = n2[i], nj = n2[j], nij = ni + nj;
  for (int d = lane; d < D; d += 32) {
    mui[d] = mu2[i * D + d];
    muj[d] = mu2[j * D + d];
    ejs[d] = ej[d];
  }
  if (lane == 0) fail_s = 0;
  __syncthreads();
  for (int d = lane; d < D; d += 32) muij[d] = (ni * mui[d] + nj * muj[d]) / nij;
  __syncthreads();

  const float ci = (ni - 1.0f) / ni, cj = (nj - 1.0f) / nj, inv = 1.0f / (nij - 1.0f);
  for (int idx = lane; idx < D * D; idx += 32) {
    int r = idx >> 6, cc = idx & 63;
    float si = S[(size_t)i * D * D + idx];
    if (i == ui) si = (ut == 0) ? (si + ejs[r] * ejs[cc]) : ((r == cc) ? 1.0f : 0.0f);
    float sj = S[(size_t)j * D * D + idx];
    if (j == ui) sj = (ut == 0) ? (sj + ejs[r] * ejs[cc]) : ((r == cc) ? 1.0f : 0.0f);
    float zzi = ci * si + mui[r] * mui[cc];
    float zzj = cj * sj + muj[r] * muj[cc];
    Sig[r][cc] = (zzi + zzj - muij[r] * muij[cc]) * inv;
  }
  __syncthreads();

  for (int kb = 0; kb < 4; ++kb) {
    const int k = kb * 16;
    // --- factor 16x16 diagonal block (lower triangle valid; upper = don't care)
    for (int c0 = 0; c0 < 16; ++c0) {
      if (lane == 0) {
        float dv = Sig[k + c0][k + c0];
        if (!(dv > 0.0f)) { fail_s = 1; dv = 1.0f; }
        sdiag_s = sqrtf(dv);
        Sig[k + c0][k + c0] = sdiag_s;
      }
      __syncthreads();
      float sd = sdiag_s;
      if (lane > c0 && lane < 16) Sig[k + lane][k + c0] /= sd;
      __syncthreads();
      int rem = 15 - c0;
      for (int t = lane; t < rem * rem; t += 32) {
        int r = c0 + 1 + t / rem, cc = c0 + 1 + t % rem;
        Sig[k + r][k + cc] -= Sig[k + r][k + c0] * Sig[k + cc][k + c0];
      }
      __syncthreads();
    }
    // --- panel triangular solve: rows k+16..63, one row per lane-slot
    for (int r = k + 16 + lane; r < D; r += 32) {
      for (int cc = 0; cc < 16; ++cc) {
        float v = Sig[r][k + cc];
        for (int p = 0; p < cc; ++p) v -= Sig[r][k + p] * Sig[k + cc][k + p];
        Sig[r][k + cc] = v / Sig[k + cc][k + cc];
      }
    }
    __syncthreads();
    // --- trailing update A22 -= P * P^T via fp32 WMMA, 16x16 tiles, K=16
    const int m = lane & 15;
    const int koff = (lane < 16) ? 0 : 2;
    for (int bi = kb + 1; bi < 4; ++bi) {
      for (int bj = kb + 1; bj <= bi; ++bj) {
        v8f acc = {};
        const int rowA = 16 * bi + m;
        const int rowB = 16 * bj + m;
        for (int kk = 0; kk < 4; ++kk) {
          v2f a, b;
          a.x = Sig[rowA][k + 4 * kk + koff];
          a.y = Sig[rowA][k + 4 * kk + koff + 1];
          b.x = Sig[rowB][k + 4 * kk + koff];
          b.y = Sig[rowB][k + 4 * kk + koff + 1];
          acc = __builtin_amdgcn_wmma_f32_16x16x4_f32(
              false, a, false, b, (short)0, acc, false, false);
        }
        // writes hit cols >= k+16 only; fragment loads read cols < k+16 -> no
        // intra-step hazard across (bi,bj) iterations.
        const int col = 16 * bj + m;
        const int rbase = 16 * bi + ((lane < 16) ? 0 : 8);
        for (int v = 0; v < 8; ++v) Sig[rbase + v][col] -= acc[v];
      }
    }
    __syncthreads();
  }

  float ld = 2.0f * (logf(Sig[lane][lane]) + logf(Sig[lane + 32][lane + 32]));
  red[lane] = ld;
  __syncthreads();
  for (int s = 16; s > 0; s >>= 1) { if (lane < s) red[lane] += red[lane + s]; __syncthreads(); }
  if (lane == 0) {
    float detij = expf(red[0]);
    kappa[f] = fail_s ? __builtin_inff() : (detij / (det_ii[i] + det_ii[j]));
  }
}

// ---------------------------------------------------------------------------
// Kernel 5: argmin(kappa) with first-index tie-break, optional merge, output.
// ---------------------------------------------------------------------------
__global__ __launch_bounds__(256) void k_final(
    const float* __restrict__ kappa, float* __restrict__ mu2,
    const float* __restrict__ n2, int* __restrict__ active2,
    float* __restrict__ out)
{
  __shared__ float mv[256];
  __shared__ int mif[256];
  __shared__ int mi_s, mj_s, merge_s;
  const int tid = threadIdx.x;

  float best = __builtin_inff();
  int bidx = 0x7fffffff;
  for (int f2 = tid; f2 < C * C; f2 += 256) {
    float v = kappa[f2];
    if (v < best || (v == best && f2 < bidx)) { best = v; bidx = f2; }
  }
  mv[tid] = best; mif[tid] = bidx;
  __syncthreads();
  for (int s = 128; s > 0; s >>= 1) {
    if (tid < s) {
      float a = mv[tid], b = mv[tid + s];
      if (b < a || (b == a && mif[tid + s] < mif[tid])) { mv[tid] = b; mif[tid] = mif[tid + s]; }
    }
    __syncthreads();
  }
  if (tid == 0) {
    float kmin = mv[0];
    int fl = mif[0];
    merge_s = (fl != 0x7fffffff && kmin < KJOIN) ? 1 : 0;
    mi_s = (fl == 0x7fffffff) ? 0 : (fl >> 7);
    mj_s = (fl == 0x7fffffff) ? 0 : (fl & 127);
  }
  __syncthreads();
  if (merge_s) {
    int mi = mi_s, mj = mj_s;
    if (tid < D) {
      float nm = n2[mi], no = n2[mj];
      mu2[mi * D + tid] = (nm * mu2[mi * D + tid] + no * mu2[mj * D + tid]) / (nm + no);
    }
    if (tid == 0) active2[mj] = 0;
  }
  __syncthreads();
  for (int idx = tid; idx < C * D; idx += 256)
    out[idx] = active2[idx >> 6] ? mu2[idx] : 0.0f;
}

// ---------------------------------------------------------------------------
extern "C" void kernel_launch(void* const* d_in, const int* in_sizes, int n_in,
                              void* d_out, int out_size, void* d_ws, size_t ws_size,
                              hipStream_t stream)
{
  const float* z = (const float*)d_in[0];     // (64,)
  const float* mu = (const float*)d_in[1];    // (128,64)
  const float* S = (const float*)d_in[2];     // (128,64,64)
  const float* n = (const float*)d_in[3];     // (128,)
  const int* c_active = (const int*)d_in[4];  // scalar
  float* out = (float*)d_out;                 // (128,64)

  float* wsf = (float*)d_ws;                  // ~101 KB used
  float* Gamma  = wsf;                        // 128
  float* mu2    = wsf + 128;                  // 8192
  float* n2     = wsf + 8320;                 // 128
  float* det_ii = wsf + 8448;                 // 128
  float* ej     = wsf + 8576;                 // 64
  float* kappa  = wsf + 8640;                 // 16384
  int*   ib     = (int*)(wsf + 25024);
  int* active2 = ib;                          // 128
  int* sel     = ib + 128;                    // 128
  int* upd     = ib + 256;                    // 2

  k_gamma<<<C, 64, 0, stream>>>(z, mu, S, n, c_active, Gamma);
  k_select<<<1, C, 0, stream>>>(z, mu, n, c_active, Gamma, mu2, n2, active2, sel, ej, upd);
  k_detii<<<C, 64, 0, stream>>>(S, n2, ej, upd, det_ii);
  k_pair<<<C * C, 32, 0, stream>>>(S, mu2, n2, sel, ej, upd, det_ii, kappa);
  k_final<<<1, 256, 0, stream>>>(kappa, mu2, n2, active2, out);

  (void)in_sizes; (void)n_in; (void)out_size; (void)ws_size;
}